// Torch_cross_correlation_81046032876249
// MI455X (gfx1250) — compile-verified
//
#include <hip/hip_runtime.h>
#include <hip/hip_bf16.h>
#include <math.h>

// ---------------- problem constants ----------------
#define NCH    1000
#define NPTS   30000
#define XCOR   59999            // 2*NPTS-1
#define FAST   65536            // nextpow2(2*NPTS-1) = 256*256
#define NF     32769            // FAST/2 + 1
#define NWIN   655              // int(0.5 / (50/65536))
#define WHALF  327              // centered window [i-327, i+327], len 655
#define IDXF1  131              // floor(0.1*65536/50)
#define IDXF2  13108            // ceil(10*65536/50)
#define PI_F     3.14159265358979323846f
#define TWOPI_F  6.28318530717958647693f

typedef float v2f __attribute__((ext_vector_type(2)));
typedef float v8f __attribute__((ext_vector_type(8)));

// ---------------- WMMA helper: D(16x16,f32) += A(16x4,f32) x B(4x16,f32) ----
static __device__ inline v8f wmma4(v2f a, v2f b, v8f c) {
  return __builtin_amdgcn_wmma_f32_16x16x4_f32(
      false, a, false, b, (short)0, c, false, false);
}

// Constant DFT-16 matrix fragments. W16 is symmetric and the f32 A-layout
// (row=lane&15, K=4c+2*hi+j) and our B-layout (col=lane&15, K=4c+j+2*hi)
// address the same element, so one fragment set serves both operand roles.
struct W16Frags {
  v2f re[4];   // Re W16 chunk c
  v2f im[4];   // Im W16 chunk c
  v2f imn[4];  // -Im W16 chunk c  (f32 WMMA has no A/B negate -> pre-negate)
};

static __device__ inline void init_w16(W16Frags& w, int lane) {
  const int hi = lane >> 4;
  const int n  = lane & 15;
#pragma unroll
  for (int c = 0; c < 4; ++c) {
#pragma unroll
    for (int j = 0; j < 2; ++j) {
      const int K = 4 * c + j + 2 * hi;            // 0..15
      const float ang = -TWOPI_F * (float)((K * n) & 15) * (1.0f / 16.0f);
      float s, cn;
      __sincosf(ang, &s, &cn);
      w.re[c][j]  = cn;
      w.im[c][j]  = s;
      w.imn[c][j] = -s;
    }
  }
}

// LDS byte offset of a __shared__ object (generic shared pointer keeps the
// workgroup-relative LDS offset in its low 32 bits on gfx1250).
static __device__ inline unsigned lds_offset_of(const void* p) {
  return (unsigned)(unsigned long long)p;
}

// 256-point complex FFT of one wave's interleaved tile in LDS (y[t] = tl[t]).
// Decomposition t = t1 + 16*t2, u = u2 + 16*u1:
//   stage1 G = M x W16 (data as A-operand), twiddle W256^{t2*u1},
//   stage2 R = W16 x H (data as B-operand).  Result in C/D layout:
//   vgpr j, lane l -> element (u2 = j + 8*(l>>4), u1 = l&15).
static __device__ inline void fft256(float2* tl, int lane,
                                     const W16Frags& w,
                                     v8f& outRe, v8f& outIm) {
  const int hi  = lane >> 4;
  const int col = lane & 15;

  // ---- stage 1: G = M * W16 ----
  v8f gre = {}; v8f gim = {};
#pragma unroll
  for (int c = 0; c < 4; ++c) {
    const int k0   = 4 * c + 2 * hi;   // t1 pair base (A layout)
    const int base = 16 * col;         // row t2 = lane&15 ; y index = t1+16*t2
    const float2 e0 = tl[k0 + base];
    const float2 e1 = tl[k0 + 1 + base];
    v2f are, aim;
    are[0] = e0.x; are[1] = e1.x;
    aim[0] = e0.y; aim[1] = e1.y;
    gre = wmma4(are, w.re[c],  gre);   // +Ar*Wr
    gre = wmma4(aim, w.imn[c], gre);   // -Ai*Wi
    gim = wmma4(are, w.im[c],  gim);   // +Ar*Wi
    gim = wmma4(aim, w.re[c],  gim);   // +Ai*Wr
  }

  // ---- twiddle W256^{t2*u1} elementwise in D-layout ----
#pragma unroll
  for (int j = 0; j < 8; ++j) {
    const int t2 = j + 8 * hi;
    const float ang = -TWOPI_F * (float)(t2 * col) * (1.0f / 256.0f);
    float s, cn;
    __sincosf(ang, &s, &cn);
    const float r = gre[j], q = gim[j];
    gre[j] = r * cn - q * s;
    gim[j] = r * s + q * cn;
  }

  __syncthreads();                      // tile reads done before overwrite
#pragma unroll
  for (int j = 0; j < 8; ++j) {
    const int t2 = j + 8 * hi;
    float2 hv;
    hv.x = gre[j];
    hv.y = gim[j];
    tl[t2 * 16 + col] = hv;             // H at [t2*16 + u1]
  }
  __syncthreads();

  // ---- stage 2: R = W16 * H ----
  v8f rre = {}; v8f rim = {};
#pragma unroll
  for (int c = 0; c < 4; ++c) {
    v2f hre, him;
#pragma unroll
    for (int j = 0; j < 2; ++j) {
      const int K = 4 * c + j + 2 * hi;   // = t2 (B layout)
      const float2 hv = tl[K * 16 + col];
      hre[j] = hv.x;
      him[j] = hv.y;
    }
    rre = wmma4(w.re[c],  hre, rre);
    rre = wmma4(w.imn[c], him, rre);
    rim = wmma4(w.re[c],  him, rim);
    rim = wmma4(w.im[c],  hre, rim);
  }
  outRe = rre;
  outIm = rim;
}

// ---------------- four-step FFT, pass 1 ----------------
// N = 65536 = 256*256, n = n1 + 256*n2, k = 256*k1 + k2.
// For each n1: F[n1][k2] = FFT256_{n2}( x[n1 + 256*n2] ) * W_N^{n1*k2},
// stored transposed at stage[ch][k2*256 + n1] so pass 2 reads contiguously.
template <bool REAL_IN>
__global__ __launch_bounds__(256) void fourstep_stage1(
    const float* __restrict__ xreal,    // [NCH][NPTS]  (REAL_IN)
    const float2* __restrict__ xcplx,   // [NCH][FAST]  (!REAL_IN)
    float2* __restrict__ stage)         // [NCH][FAST]  as [k2*256 + n1]
{
  __shared__ float2 tile[8][256];
  const int tid  = threadIdx.x;
  const int wave = tid >> 5;
  const int lane = tid & 31;
  const int ch   = blockIdx.x >> 5;                     // 32 blocks / channel
  const int n1   = ((blockIdx.x & 31) << 3) + wave;     // 0..255

  W16Frags w;
  init_w16(w, lane);
  float2* tl = tile[wave];

  if (REAL_IN) {
    // zero-padded, predicated gather: y[t] = x[n1 + 256*t], t in [0,256)
#pragma unroll
    for (int it = 0; it < 8; ++it) {
      const int t = it * 32 + lane;
      const long g = (long)n1 + 256L * t;               // < FAST
      float2 v;
      v.x = (g < (long)NPTS) ? xreal[(long)ch * NPTS + g] : 0.0f;
      v.y = 0.0f;
      tl[t] = v;
    }
  } else {
    // in-range strided gather: async global->LDS (ASYNCcnt path), one 64-bit
    // complex element per lane per issue, no VGPR staging.
    const unsigned lbase = lds_offset_of(tl);
#pragma unroll
    for (int it = 0; it < 8; ++it) {
      const int t = it * 32 + lane;
      const float2* gp = xcplx + (long)ch * FAST + (long)n1 + 256L * t;
      const unsigned loff = lbase + (unsigned)t * 8u;
      asm volatile("global_load_async_to_lds_b64 %0, %1, off"
                   :: "v"(loff), "v"(gp) : "memory");
    }
    asm volatile("s_wait_asynccnt 0x0" ::: "memory");
  }
  __syncthreads();

  v8f yre, yim;
  fft256(tl, lane, w, yre, yim);

  const int hi = lane >> 4, col = lane & 15;
#pragma unroll
  for (int j = 0; j < 8; ++j) {
    const int u = (j + 8 * hi) + 16 * col;              // k2
    // four-step twiddle W_N^{n1*k2}; n1*u <= 255*255 < 65536
    const float ang = -TWOPI_F * (float)(n1 * u) * (1.0f / (float)FAST);
    float s, cn;
    __sincosf(ang, &s, &cn);
    const float r = yre[j], q = yim[j];
    float2 o;
    o.x = r * cn - q * s;
    o.y = r * s + q * cn;
    stage[(long)ch * FAST + (long)u * 256 + n1] = o;
  }
}

// ---------------- four-step FFT, pass 2 ----------------
// For each k2: X[256*k1 + k2] = FFT256_{n1}( stage[k2*256 + n1] ).
// MODE 0: write full complex spectrum.  MODE 1: final inverse pass — write
// real part / N into the rolled+cropped output position
// j = (m + 29999) mod 65536, kept iff j < 59999.
template <int MODE>
__global__ __launch_bounds__(256) void fourstep_stage2(
    const float2* __restrict__ stage,   // [NCH][k2*256 + n1]
    float2* __restrict__ spec,          // [NCH][FAST]   (MODE 0)
    float* __restrict__ out)            // [NCH][XCOR]   (MODE 1)
{
  __shared__ float2 tile[8][256];
  const int tid  = threadIdx.x;
  const int wave = tid >> 5;
  const int lane = tid & 31;
  const int ch   = blockIdx.x >> 5;
  const int k2   = ((blockIdx.x & 31) << 3) + wave;

  W16Frags w;
  init_w16(w, lane);
  float2* tl = tile[wave];

  // contiguous 2KB tile per wave: stream straight into LDS (ASYNCcnt path)
  const float2* src = stage + (long)ch * FAST + (long)k2 * 256;
  const unsigned lbase = lds_offset_of(tl);
#pragma unroll
  for (int it = 0; it < 8; ++it) {
    const int t = it * 32 + lane;
    const float2* gp = src + t;
    const unsigned loff = lbase + (unsigned)t * 8u;
    asm volatile("global_load_async_to_lds_b64 %0, %1, off"
                 :: "v"(loff), "v"(gp) : "memory");
  }
  asm volatile("s_wait_asynccnt 0x0" ::: "memory");
  __syncthreads();

  v8f yre, yim;
  fft256(tl, lane, w, yre, yim);

  const int hi = lane >> 4, col = lane & 15;
#pragma unroll
  for (int j = 0; j < 8; ++j) {
    const int u = (j + 8 * hi) + 16 * col;              // k1
    const long k = 256L * u + k2;
    if (MODE == 0) {
      float2 o;
      o.x = yre[j];
      o.y = yim[j];
      spec[(long)ch * FAST + k] = o;
    } else {
      long jdx = k + (long)(NPTS - 1);                  // m + 29999
      if (jdx >= (long)FAST) jdx -= (long)FAST;
      if (jdx < (long)XCOR)
        out[(long)ch * XCOR + jdx] = yre[j] * (1.0f / (float)FAST);
    }
  }
}

// ---------------- whitening: RAM smoothing + tapers + Hermitian mirror -----
// One block per channel. Phase 1: inclusive cumsum of a[k]=|S[k]| (k<NF) via
// chunked Hillis-Steele scan with carry. Phase 2: scipy-'same' box filter
// with edge replication reduces to clamped window [i-327, i+327]; divide,
// nan_to_num, cos^2 tapers, rebuild full Hermitian spectrum in place.
__global__ __launch_bounds__(256) void whiten_kernel(
    float2* __restrict__ spec,          // [NCH][FAST] in/out
    float* __restrict__ csbuf)          // [NCH][NF] scratch (inclusive cumsum)
{
  __shared__ float sh[256];
  __shared__ float carry_sh;
  const int ch  = blockIdx.x;
  const int tid = threadIdx.x;
  float2* S   = spec + (long)ch * FAST;
  float*  CSI = csbuf + (long)ch * NF;

  if (tid == 0) carry_sh = 0.0f;
  __syncthreads();

  const int nchunk = (NF + 255) / 256;  // 129
  for (int cidx = 0; cidx < nchunk; ++cidx) {
    const int i = cidx * 256 + tid;
    float v = 0.0f;
    if (i < NF) {
      const float2 z = S[i];
      v = sqrtf(z.x * z.x + z.y * z.y);
    }
    sh[tid] = v;
    __syncthreads();
#pragma unroll
    for (int off = 1; off < 256; off <<= 1) {
      float t = sh[tid];
      if (tid >= off) t += sh[tid - off];
      __syncthreads();
      sh[tid] = t;
      __syncthreads();
    }
    const float c0 = carry_sh;
    if (i < NF) CSI[i] = sh[tid] + c0;
    __syncthreads();
    if (tid == 255) carry_sh = c0 + sh[255];
    __syncthreads();
  }

  const float2 z0 = S[0];
  const float2 zL = S[NF - 1];
  const float a0 = sqrtf(z0.x * z0.x + z0.y * z0.y);
  const float aL = sqrtf(zL.x * zL.x + zL.y * zL.y);

  for (int i = tid; i < NF; i += 256) {
    int lo = i - WHALF, below = 0;
    if (lo < 0) { below = -lo; lo = 0; }
    int hiW = i + WHALF, above = 0;
    if (hiW > NF - 1) { above = hiW - (NF - 1); hiW = NF - 1; }
    float s = CSI[hiW] - (lo > 0 ? CSI[lo - 1] : 0.0f);
    s += (float)below * a0 + (float)above * aL;
    const float sm  = s * (1.0f / (float)NWIN);
    const float inv = 1.0f / sm;
    const float2 z = S[i];
    float wr = z.x * inv, wi = z.y * inv;
    if (!isfinite(wr)) wr = 0.0f;
    if (!isfinite(wi)) wi = 0.0f;

    float t = 1.0f;
    if (i < IDXF1) {
      const float ang = 0.5f * PI_F + 0.5f * PI_F * (float)i / (float)(IDXF1 - 1);
      const float cv = __cosf(ang);
      t = cv * cv;
    } else if (i >= IDXF2) {
      const int jj = i - IDXF2;
      const float ang = PI_F - 0.5f * PI_F * (float)jj / (float)(NF - IDXF2 - 1);
      const float cv = __cosf(ang);
      t = cv * cv;
    }
    wr *= t;
    wi *= t;

    float2 wv; wv.x = wr; wv.y = wi;
    S[i] = wv;
    if (i > 0 && i < FAST / 2) {        // Hermitian mirror (writes > NF only)
      float2 mc; mc.x = wr; mc.y = -wi;
      S[FAST - i] = mc;
    }
  }
}

// ---------------- cross spectrum, pre-conjugated for inverse-via-forward ---
// irfft(conj(F1)*F2)[n] = Re( FFT( F1 * conj(F2) )[n] ) / N
__global__ __launch_bounds__(256) void cross_kernel(
    const float2* __restrict__ w1,
    const float2* __restrict__ w2,
    float2* __restrict__ p, long n)
{
  const long i = (long)blockIdx.x * blockDim.x + threadIdx.x;
  if (i >= n) return;
  const float2 a = w1[i];
  const float2 b = w2[i];
  float2 r;
  r.x = a.x * b.x + a.y * b.y;          // a * conj(b)
  r.y = a.y * b.x - a.x * b.y;
  p[i] = r;
}

// ---------------- host-side orchestration ----------------
extern "C" void kernel_launch(void* const* d_in, const int* in_sizes, int n_in,
                              void* d_out, int out_size, void* d_ws, size_t ws_size,
                              hipStream_t stream) {
  const float* d1 = (const float*)d_in[0];
  const float* d2 = (const float*)d_in[1];
  float* out = (float*)d_out;

  // workspace carve-up (~1.70 GB):
  //   stage  : NCH*FAST complex  (524 MB, reused per transform)
  //   spec1/2: NCH*FAST complex  (524 MB each)
  //   csbuf  : NCH*NF float      (131 MB)
  const size_t nSpec = (size_t)NCH * FAST;
  float2* stage = (float2*)d_ws;
  float2* spec1 = stage + nSpec;
  float2* spec2 = spec1 + nSpec;
  float*  cs    = (float*)(spec2 + nSpec);

  const dim3 blk(256);
  const dim3 grdFFT(NCH * 32);          // 32 blocks (256 waves) per channel

  // forward FFT + spectrum, signal 1
  fourstep_stage1<true><<<grdFFT, blk, 0, stream>>>(d1, nullptr, stage);
  fourstep_stage2<0><<<grdFFT, blk, 0, stream>>>(stage, spec1, nullptr);
  // forward FFT + spectrum, signal 2
  fourstep_stage1<true><<<grdFFT, blk, 0, stream>>>(d2, nullptr, stage);
  fourstep_stage2<0><<<grdFFT, blk, 0, stream>>>(stage, spec2, nullptr);

  // spectral whitening (in place, Hermitian-extended)
  whiten_kernel<<<dim3(NCH), blk, 0, stream>>>(spec1, cs);
  whiten_kernel<<<dim3(NCH), blk, 0, stream>>>(spec2, cs);

  // cross spectrum P = W1 * conj(W2)   (in place over spec1)
  const long nTot = (long)NCH * FAST;
  cross_kernel<<<dim3((unsigned)((nTot + 255) / 256)), blk, 0, stream>>>(
      spec1, spec2, spec1, nTot);

  // inverse FFT via forward transform of conjugated product; final pass folds
  // roll(FAST/2) + center crop into the store index.
  fourstep_stage1<false><<<grdFFT, blk, 0, stream>>>(nullptr, (const float2*)spec1, stage);
  fourstep_stage2<1><<<grdFFT, blk, 0, stream>>>(stage, nullptr, out);
}